// GATLayer_63136019251381
// MI455X (gfx1250) — compile-verified
//
#include <hip/hip_runtime.h>

typedef __attribute__((ext_vector_type(2))) float v2f;
typedef __attribute__((ext_vector_type(8))) float v8f;

#define LEAKY   0.2f
#define EPSF    1e-10f
#define IN_F    128
#define OUT_F   64
#define MAXBLK  1024

// ---------------------------------------------------------------------------
// Kernel 1: Wh = h @ W via V_WMMA_F32_16X16X4_F32, fused f1 = Wh@a1, f2 = Wh@a2.
// 256 threads = 8 waves; wave -> one 16x16 tile: 2 row-tiles x 4 col-tiles/block.
// W (128x64 = 32KB) staged in LDS, coalesced.
// A fragment layout (ISA 7.12.2, 32-bit A 16x4): lanes 0-15 rows M=0..15 with
// K={0,1} in v[0],v[1]; lanes 16-31 same rows with K={2,3}.
// C/D layout: lanes 0-15 -> M=r (vgpr r), N=lane; lanes 16-31 -> M=8+r, N=lane-16.
// ---------------------------------------------------------------------------
__global__ void __launch_bounds__(256) gat_gemm_wmma(
    const float* __restrict__ h, const float* __restrict__ W,
    const float* __restrict__ a1, const float* __restrict__ a2,
    float* __restrict__ Wh, float* __restrict__ f1, float* __restrict__ f2,
    int N)
{
    __shared__ float sW[IN_F * OUT_F];
    for (int i = threadIdx.x; i < IN_F * OUT_F; i += 256) sW[i] = W[i];
    __syncthreads();

    const int lane    = threadIdx.x & 31;
    const int wave    = threadIdx.x >> 5;
    const int rowTile = blockIdx.x * 2 + (wave >> 2);
    const int colTile = wave & 3;
    const int row0    = rowTile * 16;
    if (row0 >= N) return;                        // wave-uniform; EXEC stays all-1s

    const int mh = lane & 15;                     // A-row / B-col within tile
    const int kh = lane >> 4;                     // K sub-pair select (0 or 1)
    int rowA = row0 + mh; if (rowA >= N) rowA = N - 1;   // clamp (reads only)
    const float* __restrict__ hrow = h + (size_t)rowA * IN_F;
    const int n = colTile * 16 + mh;              // global output column

    v8f c = {};
    for (int k0 = 0; k0 < IN_F; k0 += 4) {
        const int k = k0 + 2 * kh;
        v2f a; a[0] = hrow[k];              a[1] = hrow[k + 1];
        v2f b; b[0] = sW[k * OUT_F + n];    b[1] = sW[(k + 1) * OUT_F + n];
        c = __builtin_amdgcn_wmma_f32_16x16x4_f32(
                /*neg_a=*/false, a, /*neg_b=*/false, b,
                /*c_mod=*/(short)0, c, /*reuse_a=*/false, /*reuse_b=*/false);
    }

    const float a1n = a1[n];
    const float a2n = a2[n];
#pragma unroll
    for (int r = 0; r < 8; ++r) {
        const int m    = kh ? (8 + r) : r;
        const int grow = row0 + m;
        const float v  = c[r];
        if (grow < N) Wh[(size_t)grow * OUT_F + n] = v;
        float t1 = v * a1n;
        float t2 = v * a2n;
#pragma unroll
        for (int o = 8; o >= 1; o >>= 1) {        // reduce across the 16 columns
            t1 += __shfl_xor(t1, o, 16);
            t2 += __shfl_xor(t2, o, 16);
        }
        if (mh == 0 && grow < N) {
            atomicAdd(&f1[grow], t1);
            atomicAdd(&f2[grow], t2);
        }
    }
}

// ---------------------------------------------------------------------------
// Kernel 2: per-block max of e = leaky_relu(f1[row] + f2[col])
// ---------------------------------------------------------------------------
__global__ void __launch_bounds__(256) gat_edge_max(
    const int* __restrict__ ei, const float* __restrict__ f1,
    const float* __restrict__ f2, float* __restrict__ blockmax, int E)
{
    float m = -3.402823466e+38f;
    for (int i = blockIdx.x * 256 + threadIdx.x; i < E; i += gridDim.x * 256) {
        const int r = ei[i], cidx = ei[E + i];
        const float x = f1[r] + f2[cidx];
        const float e = x > 0.f ? x : LEAKY * x;
        m = fmaxf(m, e);
    }
#pragma unroll
    for (int o = 16; o >= 1; o >>= 1) m = fmaxf(m, __shfl_xor(m, o, 32));
    __shared__ float sm[8];
    if ((threadIdx.x & 31) == 0) sm[threadIdx.x >> 5] = m;
    __syncthreads();
    if (threadIdx.x == 0) {
        float g = sm[0];
#pragma unroll
        for (int i = 1; i < 8; ++i) g = fmaxf(g, sm[i]);
        blockmax[blockIdx.x] = g;
    }
}

// Kernel 3: reduce block maxima -> single scalar
__global__ void __launch_bounds__(256) gat_reduce_max(
    const float* __restrict__ blockmax, float* __restrict__ gmax, int n)
{
    float m = -3.402823466e+38f;
    for (int i = threadIdx.x; i < n; i += 256) m = fmaxf(m, blockmax[i]);
#pragma unroll
    for (int o = 16; o >= 1; o >>= 1) m = fmaxf(m, __shfl_xor(m, o, 32));
    __shared__ float sm[8];
    if ((threadIdx.x & 31) == 0) sm[threadIdx.x >> 5] = m;
    __syncthreads();
    if (threadIdx.x == 0) {
        float g = sm[0];
#pragma unroll
        for (int i = 1; i < 8; ++i) g = fmaxf(g, sm[i]);
        *gmax = g;
    }
}

// ---------------------------------------------------------------------------
// Kernel 4: e_exp[i] = exp(e - gmax); exp_sum[row] += e_exp[i]
// ---------------------------------------------------------------------------
__global__ void __launch_bounds__(256) gat_edge_exp(
    const int* __restrict__ ei, const float* __restrict__ f1,
    const float* __restrict__ f2, const float* __restrict__ gmax,
    float* __restrict__ e_exp, float* __restrict__ exp_sum, int E)
{
    const float g = *gmax;
    for (int i = blockIdx.x * 256 + threadIdx.x; i < E; i += gridDim.x * 256) {
        const int r = ei[i], cidx = ei[E + i];
        const float x  = f1[r] + f2[cidx];
        const float e  = x > 0.f ? x : LEAKY * x;
        const float ex = __expf(e - g);
        e_exp[i] = ex;
        atomicAdd(&exp_sum[r], ex);
    }
}

// ---------------------------------------------------------------------------
// Kernel 5: out[row] += Wh[col] * alpha    (one wave per edge; lane handles
// elements lane and lane+32 -> two 128B coalesced gathers + 64 f32 atomics)
// ---------------------------------------------------------------------------
__global__ void __launch_bounds__(256) gat_scatter(
    const int* __restrict__ ei, const float* __restrict__ Wh,
    const float* __restrict__ e_exp, const float* __restrict__ exp_sum,
    float* __restrict__ out, int E)
{
    const int lane = threadIdx.x & 31;
    const int wave = threadIdx.x >> 5;
    const int i = blockIdx.x * 8 + wave;
    if (i >= E) return;
    const int r = ei[i], cidx = ei[E + i];
    const float alpha = e_exp[i] / (exp_sum[r] + EPSF);
    const float* __restrict__ src = Wh + (size_t)cidx * OUT_F;
    float* __restrict__ dst = out + (size_t)r * OUT_F;
    atomicAdd(&dst[lane],      src[lane]      * alpha);
    atomicAdd(&dst[lane + 32], src[lane + 32] * alpha);
}

// Kernel 6: in-place ELU
__global__ void __launch_bounds__(256) gat_elu(float* __restrict__ out, int n)
{
    const int i = blockIdx.x * 256 + threadIdx.x;
    if (i < n) {
        const float x = out[i];
        out[i] = x > 0.f ? x : (__expf(x) - 1.0f);
    }
}

// ---------------------------------------------------------------------------
extern "C" void kernel_launch(void* const* d_in, const int* in_sizes, int n_in,
                              void* d_out, int out_size, void* d_ws, size_t ws_size,
                              hipStream_t stream)
{
    const float* h  = (const float*)d_in[0];
    const float* W  = (const float*)d_in[1];
    const float* a1 = (const float*)d_in[2];
    const float* a2 = (const float*)d_in[3];
    const int*   ei = (const int*)  d_in[4];
    float* out = (float*)d_out;

    const int N = in_sizes[0] / IN_F;   // 100000
    const int E = in_sizes[4] / 2;      // 1600000

    // Workspace layout (floats): Wh[N*64] | f1[N] f2[N] exp_sum[N] | e_exp[E] | blockmax[MAXBLK] | gmax[1]
    float* ws       = (float*)d_ws;
    float* Wh       = ws;
    float* f1       = Wh + (size_t)N * OUT_F;
    float* f2       = f1 + N;
    float* exp_sum  = f2 + N;
    float* e_exp    = exp_sum + N;
    float* blockmax = e_exp + E;
    float* gmax     = blockmax + MAXBLK;

    // Zero the accumulators (f1|f2|exp_sum are contiguous) and the output.
    hipMemsetAsync(f1, 0, sizeof(float) * (size_t)3 * N, stream);
    hipMemsetAsync(out, 0, sizeof(float) * (size_t)out_size, stream);

    const dim3 blk(256);
    const int gemmBlocks = (N + 31) / 32;          // 2 row-tiles of 16 per block
    gat_gemm_wmma<<<gemmBlocks, blk, 0, stream>>>(h, W, a1, a2, Wh, f1, f2, N);

    gat_edge_max<<<MAXBLK, blk, 0, stream>>>(ei, f1, f2, blockmax, E);
    gat_reduce_max<<<1, blk, 0, stream>>>(blockmax, gmax, MAXBLK);
    gat_edge_exp<<<(E + 255) / 256, blk, 0, stream>>>(ei, f1, f2, gmax, e_exp, exp_sum, E);
    gat_scatter<<<(E + 7) / 8, blk, 0, stream>>>(ei, Wh, e_exp, exp_sum, out, E);
    gat_elu<<<(out_size + 255) / 256, blk, 0, stream>>>(out, out_size);
}